// MTLU_36344013259315
// MI455X (gfx1250) — compile-verified
//
#include <hip/hip_runtime.h>
#include <cstdint>

// MTLU: out[b,c,h,w] = weight[c, bin(x)] * x + bias[c, bin(x)]
// bin(x) = clamp(floor(x*20) + 20, 0, 39)
//
// Memory-bound streaming kernel (537 MB @ 23.3 TB/s ~= 23 us). No WMMA — a
// matrix formulation would only inflate traffic. CDNA5 paths used instead:
//   * global_load_async_to_lds_b32 + s_wait_asynccnt for LUT staging
//   * conflict-free LDS gather (40 fp32 entries < 64 banks, per-array)
//   * nontemporal B128 loads/stores for the streams that exceed 192MB L2

typedef float v4f __attribute__((ext_vector_type(4)));
typedef __attribute__((address_space(1))) int* gptr_i32;
typedef __attribute__((address_space(3))) int* lptr_i32;

#define BIN_NUM   40
#define HALF_BIN  20
#define INV_BIN_W 20.0f          // 1.0/0.05 rounds exactly to 20.0
#define CH        64
#define HW        65536          // 256*256 elements per (b,c) image

constexpr int TPB  = 256;        // 8 wave32s per block
constexpr int VEC  = 4;          // float4 accesses
constexpr int ITER = 4;          // float4s per thread
constexpr int ELEMS_PER_BLOCK = TPB * VEC * ITER;      // 4096
constexpr int BLOCKS_PER_IMG  = HW / ELEMS_PER_BLOCK;  // 16

__global__ __launch_bounds__(TPB) void mtlu_36344013259315_kernel(
    const float* __restrict__ x,
    const float* __restrict__ weight,
    const float* __restrict__ bias,
    float* __restrict__ out)
{
    // Separate arrays: entries 0..39 each map to a unique LDS bank, so the
    // data-dependent ds_load_b32 gather below can never bank-conflict.
    __shared__ float lds_w[BIN_NUM];
    __shared__ float lds_b[BIN_NUM];

    const int tid  = threadIdx.x;
    const int img  = blockIdx.x >> 4;                 // BLOCKS_PER_IMG == 16
    const int tile = blockIdx.x & (BLOCKS_PER_IMG - 1);
    const int c    = img & (CH - 1);                  // layout [B, C, H, W]

    // ---- stage this channel's 40-entry LUT into LDS via async global->LDS DMA
    if (tid < BIN_NUM) {
        const float* gw = weight + c * BIN_NUM + tid;
        const float* gb = bias   + c * BIN_NUM + tid;
#if __has_builtin(__builtin_amdgcn_global_load_async_to_lds_b32)
        __builtin_amdgcn_global_load_async_to_lds_b32(
            (gptr_i32)gw, (lptr_i32)&lds_w[tid], 0, 0);
        __builtin_amdgcn_global_load_async_to_lds_b32(
            (gptr_i32)gb, (lptr_i32)&lds_b[tid], 0, 0);
#else
        uint32_t lw = (uint32_t)(uintptr_t)&lds_w[tid];   // low 32 bits = LDS offset
        uint32_t lb = (uint32_t)(uintptr_t)&lds_b[tid];
        uint64_t aw = (uint64_t)(uintptr_t)gw;
        uint64_t ab = (uint64_t)(uintptr_t)gb;
        asm volatile("global_load_async_to_lds_b32 %0, %1, off\n\t"
                     "global_load_async_to_lds_b32 %2, %3, off"
                     :: "v"(lw), "v"(aw), "v"(lb), "v"(ab) : "memory");
#endif
    }
#if __has_builtin(__builtin_amdgcn_s_wait_asynccnt)
    __builtin_amdgcn_s_wait_asynccnt(0);
#else
    asm volatile("s_wait_asynccnt 0x0" ::: "memory");
#endif
    __syncthreads();

    // ---- stream 4096 elements per block as float4, NT-hinted both ways
    const long imgBase = (long)img * HW;
    const v4f* __restrict__ xin  =
        (const v4f*)(x + imgBase) + tile * (ELEMS_PER_BLOCK / VEC);
    v4f* __restrict__ outp =
        (v4f*)(out + imgBase) + tile * (ELEMS_PER_BLOCK / VEC);

#pragma unroll
    for (int i = 0; i < ITER; ++i) {
        const int p = i * TPB + tid;
        v4f v = __builtin_nontemporal_load(&xin[p]);
        v4f r;
#pragma unroll
        for (int k = 0; k < VEC; ++k) {
            float xv  = v[k];
            int   idx = (int)floorf(xv * INV_BIN_W) + HALF_BIN;
            idx = idx < 0 ? 0 : (idx > (BIN_NUM - 1) ? (BIN_NUM - 1) : idx);
            r[k] = fmaf(lds_w[idx], xv, lds_b[idx]);   // conflict-free ds_load_b32 x2
        }
        __builtin_nontemporal_store(r, &outp[p]);
    }
}

extern "C" void kernel_launch(void* const* d_in, const int* in_sizes, int n_in,
                              void* d_out, int out_size, void* d_ws, size_t ws_size,
                              hipStream_t stream) {
    (void)n_in; (void)out_size; (void)d_ws; (void)ws_size;
    const float* x  = (const float*)d_in[0];   // [B, C, H, W] fp32
    const float* w  = (const float*)d_in[1];   // [C, 40] fp32
    const float* b  = (const float*)d_in[2];   // [C, 40] fp32
    float* out      = (float*)d_out;

    const int nimg   = in_sizes[0] / HW;              // B*C = 1024
    const int blocks = nimg * BLOCKS_PER_IMG;         // 16384
    mtlu_36344013259315_kernel<<<blocks, TPB, 0, stream>>>(x, w, b, out);
}